// SinkhornDivergence_21105469292706
// MI455X (gfx1250) — compile-verified
//
#include <hip/hip_runtime.h>
#include <math.h>

// Problem constants (match reference)
#define Nn 4096
#define Dd 1024
#define EPS 0.1f
#define CLAMPV 1e-10f
#define MUV (1.0f / 4096.0f)
#define MAX_ITER 100

typedef __attribute__((ext_vector_type(2))) float v2f;
typedef __attribute__((ext_vector_type(8))) float v8f;

// ---------------- GEMM tiling ----------------
#define BM 128
#define BN 128
#define BK 32
#define LDP 34   // padded LDS row stride (floats): conflict-free, 8B aligned

// Row squared norms: one block per row, Dd/4 == 256 float4 lanes exactly.
__global__ __launch_bounds__(256)
void sdiv_rownorm(const float* __restrict__ A, float* __restrict__ sq) {
  __shared__ float red[256];
  const int i = blockIdx.x;
  const int t = threadIdx.x;
  const float4 a = reinterpret_cast<const float4*>(A + (size_t)i * Dd)[t];
  red[t] = a.x * a.x + a.y * a.y + a.z * a.z + a.w * a.w;
  __syncthreads();
  for (int off = 128; off > 0; off >>= 1) {
    if (t < off) red[t] += red[t + off];
    __syncthreads();
  }
  if (t == 0) sq[i] = red[0];
}

// K[i,j] = exp(-max(|x_i|^2 + |y_j|^2 - 2 x_i.y_j, 0) / EPS)
// 128x128 block tile, 8 waves (2x4), each wave: 4x2 tiles of 16x16 via
// V_WMMA_F32_16X16X4_F32 with LDS-staged fragments.
__global__ __launch_bounds__(256)
void sdiv_gemm_exp(const float* __restrict__ A, const float* __restrict__ B,
                   const float* __restrict__ sqA, const float* __restrict__ sqB,
                   float* __restrict__ Km) {
  __shared__ float As[BM * LDP];
  __shared__ float Bs[BN * LDP];
  const int tid    = threadIdx.x;
  const int lane   = tid & 31;
  const int wave   = tid >> 5;     // 0..7
  const int waveM  = wave >> 2;    // 0..1 -> 64-row strip
  const int waveN  = wave & 3;     // 0..3 -> 32-col strip
  const int lane16 = lane & 15;
  const int hi     = lane >> 4;    // 0: K=0,1   1: K=2,3 (A/B frag layout)
  const int blockM = blockIdx.y * BM;
  const int blockN = blockIdx.x * BN;

  v8f acc[4][2] = {};

  const int ldRow = tid >> 3;         // 0..31
  const int ldCol = (tid & 7) << 2;   // 0,4,...,28

  for (int kb = 0; kb < Dd; kb += BK) {
    #pragma unroll
    for (int rr = 0; rr < BM; rr += 32) {
      const int r = ldRow + rr;
      const float4 av = *reinterpret_cast<const float4*>(
          &A[(size_t)(blockM + r) * Dd + kb + ldCol]);
      As[r * LDP + ldCol + 0] = av.x;
      As[r * LDP + ldCol + 1] = av.y;
      As[r * LDP + ldCol + 2] = av.z;
      As[r * LDP + ldCol + 3] = av.w;
      const float4 bv = *reinterpret_cast<const float4*>(
          &B[(size_t)(blockN + r) * Dd + kb + ldCol]);
      Bs[r * LDP + ldCol + 0] = bv.x;
      Bs[r * LDP + ldCol + 1] = bv.y;
      Bs[r * LDP + ldCol + 2] = bv.z;
      Bs[r * LDP + ldCol + 3] = bv.w;
    }
    __syncthreads();

    #pragma unroll
    for (int kk = 0; kk < BK; kk += 4) {
      const int kidx = kk + (hi << 1);
      v2f a[4], b[2];
      #pragma unroll
      for (int tm = 0; tm < 4; ++tm)
        a[tm] = *reinterpret_cast<const v2f*>(
            &As[(waveM * 64 + tm * 16 + lane16) * LDP + kidx]);
      #pragma unroll
      for (int tn = 0; tn < 2; ++tn)
        b[tn] = *reinterpret_cast<const v2f*>(
            &Bs[(waveN * 32 + tn * 16 + lane16) * LDP + kidx]);
      #pragma unroll
      for (int tm = 0; tm < 4; ++tm)
        #pragma unroll
        for (int tn = 0; tn < 2; ++tn)
          acc[tm][tn] = __builtin_amdgcn_wmma_f32_16x16x4_f32(
              false, a[tm], false, b[tn], (short)0, acc[tm][tn], false, false);
    }
    __syncthreads();
  }

  // Epilogue: C/D layout -> lane (hi,lane16), VGPR r => M = r + 8*hi, N = lane16
  #pragma unroll
  for (int tm = 0; tm < 4; ++tm) {
    const int gm0 = blockM + waveM * 64 + tm * 16 + (hi << 3);
    #pragma unroll
    for (int tn = 0; tn < 2; ++tn) {
      const int gn = blockN + waveN * 32 + tn * 16 + lane16;
      const float sb = sqB[gn];
      #pragma unroll
      for (int r = 0; r < 8; ++r) {
        const int gm = gm0 + r;
        float c = sqA[gm] + sb - 2.0f * acc[tm][tn][r];
        c = fmaxf(c, 0.0f);
        Km[(size_t)gm * Nn + gn] = __expf(-c * (1.0f / EPS));
      }
    }
  }
}

// u = v = 1
__global__ __launch_bounds__(256)
void sdiv_init_uv(float* __restrict__ uv) {
  const int i = blockIdx.x * 256 + threadIdx.x;
  if (i < 2 * Nn) uv[i] = 1.0f;
}

// u[i] = mu / max((K v)[i], CLAMP) ; one block per row
__global__ __launch_bounds__(256)
void sdiv_kv(const float* __restrict__ Km, const float* __restrict__ v,
             float* __restrict__ u) {
  __shared__ float red[256];
  const int i = blockIdx.x;
  const int t = threadIdx.x;
  const float4* row4 = reinterpret_cast<const float4*>(Km + (size_t)i * Nn);
  const float4* v4   = reinterpret_cast<const float4*>(v);
  float s = 0.0f;
  #pragma unroll
  for (int j = t; j < Nn / 4; j += 256) {
    const float4 k = row4[j];
    const float4 w = v4[j];
    s += k.x * w.x + k.y * w.y + k.z * w.z + k.w * w.w;
  }
  red[t] = s;
  __syncthreads();
  for (int off = 128; off > 0; off >>= 1) {
    if (t < off) red[t] += red[t + off];
    __syncthreads();
  }
  if (t == 0) u[i] = MUV / fmaxf(red[0], CLAMPV);
}

// v[j] = nu / max((K^T u)[j], CLAMP) ; 32 cols per block, one wave per row-group
__global__ __launch_bounds__(256)
void sdiv_ktu(const float* __restrict__ Km, const float* __restrict__ u,
              float* __restrict__ v) {
  __shared__ float red[256];
  const int lane = threadIdx.x & 31;
  const int g    = threadIdx.x >> 5;      // 0..7
  const int col  = blockIdx.x * 32 + lane;
  float s = 0.0f;
  for (int i = g; i < Nn; i += 8)
    s += Km[(size_t)i * Nn + col] * u[i];
  red[threadIdx.x] = s;
  __syncthreads();
  if (threadIdx.x < 32) {
    float t = red[lane];
    #pragma unroll
    for (int gg = 1; gg < 8; ++gg) t += red[gg * 32 + lane];
    v[col] = MUV / fmaxf(t, CLAMPV);
  }
}

// partial[i] = sum_j u_i * K_ij * v_j * C_ij with C_ij = -EPS * ln(K_ij)
__global__ __launch_bounds__(256)
void sdiv_cost_partial(const float* __restrict__ Km, const float* __restrict__ u,
                       const float* __restrict__ v, float* __restrict__ partial) {
  __shared__ float red[256];
  const int i = blockIdx.x;
  const int t = threadIdx.x;
  const float ui = u[i];
  const float* row = Km + (size_t)i * Nn;
  float s = 0.0f;
  for (int j = t; j < Nn; j += 256) {
    const float k = row[j];
    if (k > 0.0f) s += ui * v[j] * k * (-EPS * __logf(k));
  }
  red[t] = s;
  __syncthreads();
  for (int off = 128; off > 0; off >>= 1) {
    if (t < off) red[t] += red[t + off];
    __syncthreads();
  }
  if (t == 0) partial[i] = red[0];
}

__global__ __launch_bounds__(256)
void sdiv_reduce(const float* __restrict__ partial, float* __restrict__ outS) {
  __shared__ float red[256];
  const int t = threadIdx.x;
  float s = 0.0f;
  for (int i = t; i < Nn; i += 256) s += partial[i];
  red[t] = s;
  __syncthreads();
  for (int off = 128; off > 0; off >>= 1) {
    if (t < off) red[t] += red[t + off];
    __syncthreads();
  }
  if (t == 0) *outS = red[0];
}

__global__ void sdiv_combine(const float* __restrict__ S, float* __restrict__ out) {
  out[0] = S[0] - 0.5f * (S[1] + S[2]);
}

extern "C" void kernel_launch(void* const* d_in, const int* in_sizes, int n_in,
                              void* d_out, int out_size, void* d_ws, size_t ws_size,
                              hipStream_t stream) {
  const float* X = (const float*)d_in[0];
  const float* Y = (const float*)d_in[1];
  float* out = (float*)d_out;

  float* ws   = (float*)d_ws;
  float* Km   = ws;                          // 4096*4096 = 16,777,216 floats (64 MB)
  float* sqX  = Km + (size_t)Nn * Nn;        // 4096
  float* sqY  = sqX + Nn;                    // 4096
  float* u    = sqY + Nn;                    // 4096
  float* v    = u + Nn;                      // 4096 (contiguous with u for init)
  float* part = v + Nn;                      // 4096
  float* S    = part + Nn;                   // 3

  sdiv_rownorm<<<Nn, 256, 0, stream>>>(X, sqX);
  sdiv_rownorm<<<Nn, 256, 0, stream>>>(Y, sqY);

  const float* mA[3]  = {X,   X,   Y};
  const float* mB[3]  = {Y,   X,   Y};
  const float* msa[3] = {sqX, sqX, sqY};
  const float* msb[3] = {sqY, sqX, sqY};

  const dim3 gemmGrid(Nn / BN, Nn / BM);  // 32 x 32

  for (int m = 0; m < 3; ++m) {
    sdiv_gemm_exp<<<gemmGrid, 256, 0, stream>>>(mA[m], mB[m], msa[m], msb[m], Km);
    sdiv_init_uv<<<(2 * Nn + 255) / 256, 256, 0, stream>>>(u);
    for (int it = 0; it < MAX_ITER; ++it) {
      sdiv_kv<<<Nn, 256, 0, stream>>>(Km, v, u);
      sdiv_ktu<<<Nn / 32, 256, 0, stream>>>(Km, u, v);
    }
    sdiv_cost_partial<<<Nn, 256, 0, stream>>>(Km, u, v, part);
    sdiv_reduce<<<1, 256, 0, stream>>>(part, &S[m]);
  }
  sdiv_combine<<<1, 1, 0, stream>>>(S, out);
}